// Q_Fun_66486093742347
// MI455X (gfx1250) — compile-verified
//
#include <hip/hip_runtime.h>
#include <hip/hip_bf16.h>

typedef __attribute__((ext_vector_type(2))) float v2f;
typedef __attribute__((ext_vector_type(8))) float v8f;

#define BB 4
#define NN 10000
#define EE 160000
#define IN_DIM 64
#define HID 128
#define BN (BB * NN)          // 40000 nodes total, divisible by 16
#define NTILES (BN / 16)      // 2500

// ---------------------------------------------------------------------------
// zero a float region
__global__ void zero_kernel(float* __restrict__ p, int n) {
    int i = blockIdx.x * blockDim.x + threadIdx.x;
    if (i < n) p[i] = 0.0f;
}

// ---------------------------------------------------------------------------
// per-node scalars from edges: deg (count under j), s_pos = sum relu(w),
// s_neg = sum relu(-w), segmented by j = edge_index[:,:,1]
__global__ void edge_scal_kernel(const int* __restrict__ edge_index,   // B*E*2
                                 const float* __restrict__ edge_w,     // B*E
                                 float* __restrict__ deg,
                                 float* __restrict__ spos,
                                 float* __restrict__ sneg) {
    int i = blockIdx.x * blockDim.x + threadIdx.x;
    if (i >= BB * EE) return;
    int b = i / EE;
    int j = edge_index[2 * i + 1];
    size_t idx = (size_t)b * NN + (size_t)j;
    float w = edge_w[i];
    atomicAdd(&deg[idx], 1.0f);
    atomicAdd(&spos[idx], fmaxf(w, 0.0f));
    atomicAdd(&sneg[idx], fmaxf(-w, 0.0f));
}

// ---------------------------------------------------------------------------
// u[h] = sum_k W3[h,k]*relu(W4[k]);  v[h] = sum_k W3[h,k]*relu(-W4[k])
__global__ void uv_kernel(const float* __restrict__ W3,   // HID x HID
                          const float* __restrict__ W4,   // HID
                          float* __restrict__ u,
                          float* __restrict__ v) {
    int h = threadIdx.x;   // 128 threads
    float su = 0.0f, sv = 0.0f;
    for (int k = 0; k < HID; ++k) {
        float w3 = W3[h * HID + k];
        float w4 = W4[k];
        su += w3 * fmaxf(w4, 0.0f);
        sv += w3 * fmaxf(-w4, 0.0f);
    }
    u[h] = su;
    v[h] = sv;
}

// ---------------------------------------------------------------------------
// One S2V layer, folded form:
//   mu_out[n,h] = relu( x[n]*W1[h] + deg[n]*dot(mu_in[n,:],W2[h,:])
//                       + spos[n]*u[h] + sneg[n]*v[h] )
// Block = 256 threads (8 wave32). Block tile = 16 nodes x 128 channels.
// Wave w owns channel tile [16w,16w+16); WMMA f32 16x16x4 over K=Din.
__global__ void s2v_layer_kernel(const float* __restrict__ mu_in, int Din,
                                 float* __restrict__ mu_out,
                                 const float* __restrict__ W2,    // HID x Din
                                 const float* __restrict__ W1,    // HID
                                 const float* __restrict__ u,     // HID
                                 const float* __restrict__ v,     // HID
                                 const float* __restrict__ x,     // BN
                                 const float* __restrict__ deg,
                                 const float* __restrict__ spos,
                                 const float* __restrict__ sneg) {
    __shared__ float Atile[16 * HID];   // 16 nodes x Din (Din<=128)
    __shared__ float snode[4 * 16];     // x, deg, spos, sneg per node

    const int node0 = blockIdx.x * 16;
    const int tid = threadIdx.x;

    // A tile is 16 consecutive rows -> one contiguous block of 16*Din floats
    for (int i = tid; i < 16 * Din; i += 256)
        Atile[i] = mu_in[(size_t)node0 * Din + i];
    if (tid < 16) {
        snode[tid]      = x[node0 + tid];
        snode[16 + tid] = deg[node0 + tid];
        snode[32 + tid] = spos[node0 + tid];
        snode[48 + tid] = sneg[node0 + tid];
    }
    __syncthreads();

    const int wave = tid >> 5;
    const int lane = tid & 31;
    const int half = lane >> 4;      // 0: K pair {0,1}; 1: K pair {2,3}
    const int l16  = lane & 15;
    const int h0   = wave * 16;

    v8f acc = {};
    for (int k = 0; k < Din; k += 4) {
        const int ka = k + half * 2;
        v2f a, b;
        a.x = Atile[l16 * Din + ka];
        a.y = Atile[l16 * Din + ka + 1];
        b.x = W2[(size_t)(h0 + l16) * Din + ka];
        b.y = W2[(size_t)(h0 + l16) * Din + ka + 1];
        acc = __builtin_amdgcn_wmma_f32_16x16x4_f32(
            false, a, false, b, (short)0, acc, false, false);
    }

    // epilogue: C/D layout -> VGPR g, lane<16: m=g ; lane>=16: m=g+8 ; n=l16
    const int h = h0 + l16;
    const float W1h = W1[h], uh = u[h], vh = v[h];
#pragma unroll
    for (int g = 0; g < 8; ++g) {
        const int m = g + half * 8;
        float val = snode[m] * W1h + snode[16 + m] * acc[g]
                  + snode[32 + m] * uh + snode[48 + m] * vh;
        mu_out[(size_t)(node0 + m) * HID + h] = fmaxf(val, 0.0f);
    }
}

// ---------------------------------------------------------------------------
// per-batch column sums of mu: sum_mu[b,h] = sum_n mu[b,n,h]
__global__ void rowsum_kernel(const float* __restrict__ mu,
                              float* __restrict__ sum_mu) {
    const int b = blockIdx.y;
    const int t = blockIdx.x * blockDim.x + threadIdx.x;  // 32*256 = 8192
    const int h = t & 127;
    const int chunk = t >> 7;                             // 0..63
    float s = 0.0f;
    for (int n = chunk; n < NN; n += 64)
        s += mu[((size_t)b * NN + n) * HID + h];
    atomicAdd(&sum_mu[b * HID + h], s);
}

// ---------------------------------------------------------------------------
// c[b] = sum_h relu( dot(sum_mu[b,:], W6[h,:]) ) * W5[h]
__global__ void cb_kernel(const float* __restrict__ sum_mu,
                          const float* __restrict__ W6,
                          const float* __restrict__ W5,
                          float* __restrict__ cb) {
    __shared__ float red[HID];
    const int b = blockIdx.x;
    const int h = threadIdx.x;
    float gp = 0.0f;
    for (int k = 0; k < HID; ++k)
        gp += sum_mu[b * HID + k] * W6[h * HID + k];
    red[h] = fmaxf(gp, 0.0f) * W5[h];
    __syncthreads();
    for (int s = 64; s > 0; s >>= 1) {
        if (h < s) red[h] += red[h + s];
        __syncthreads();
    }
    if (h == 0) cb[b] = red[0];
}

// ---------------------------------------------------------------------------
// out[b,n] = c[b] + sum_h relu( dot(mu[n,:], W7[h,:]) ) * W5[128+h]
// Same WMMA tiling as the layer kernel; LDS ds_add_f32 reduction over h.
__global__ void readout_kernel(const float* __restrict__ mu,   // BN x HID
                               const float* __restrict__ W7,   // HID x HID
                               const float* __restrict__ W5,   // 256
                               const float* __restrict__ cb,   // B
                               float* __restrict__ out) {
    __shared__ float Atile[16 * HID];
    __shared__ float oacc[16];

    const int node0 = blockIdx.x * 16;
    const int tid = threadIdx.x;
    for (int i = tid; i < 16 * HID; i += 256)
        Atile[i] = mu[(size_t)node0 * HID + i];
    if (tid < 16) oacc[tid] = 0.0f;
    __syncthreads();

    const int wave = tid >> 5;
    const int lane = tid & 31;
    const int half = lane >> 4;
    const int l16  = lane & 15;
    const int h0   = wave * 16;

    v8f acc = {};
    for (int k = 0; k < HID; k += 4) {
        const int ka = k + half * 2;
        v2f a, b;
        a.x = Atile[l16 * HID + ka];
        a.y = Atile[l16 * HID + ka + 1];
        b.x = W7[(size_t)(h0 + l16) * HID + ka];
        b.y = W7[(size_t)(h0 + l16) * HID + ka + 1];
        acc = __builtin_amdgcn_wmma_f32_16x16x4_f32(
            false, a, false, b, (short)0, acc, false, false);
    }

    const int h = h0 + l16;
    const float w5h = W5[HID + h];
#pragma unroll
    for (int g = 0; g < 8; ++g) {
        const int m = g + half * 8;
        atomicAdd(&oacc[m], fmaxf(acc[g], 0.0f) * w5h);  // ds_add_f32
    }
    __syncthreads();
    if (tid < 16) {
        const int node = node0 + tid;
        out[node] = oacc[tid] + cb[node / NN];
    }
}

// ---------------------------------------------------------------------------
extern "C" void kernel_launch(void* const* d_in, const int* in_sizes, int n_in,
                              void* d_out, int out_size, void* d_ws, size_t ws_size,
                              hipStream_t stream) {
    // setup_inputs() dict order:
    const float* mu_in   = (const float*)d_in[0];   // (4,10000,64)
    const float* x       = (const float*)d_in[1];   // (4,10000,1)
    const float* edge_w  = (const float*)d_in[2];   // (4,160000,1)
    const int*   edge_ix = (const int*)  d_in[3];   // (4,160000,2) int32
    const float* W1_0    = (const float*)d_in[4];   // (128,1)
    const float* W2_0    = (const float*)d_in[5];   // (128,64)
    const float* W3_0    = (const float*)d_in[6];   // (128,128)
    const float* W4_0    = (const float*)d_in[7];   // (128,1)
    const float* W1_r    = (const float*)d_in[8];   // (2,128,1)
    const float* W2_r    = (const float*)d_in[9];   // (2,128,128)
    const float* W3_r    = (const float*)d_in[10];  // (2,128,128)
    const float* W4_r    = (const float*)d_in[11];  // (2,128,1)
    const float* W5      = (const float*)d_in[12];  // (1,256)
    const float* W6      = (const float*)d_in[13];  // (128,128)
    const float* W7      = (const float*)d_in[14];  // (128,128)
    float* out = (float*)d_out;                     // (4,10000)

    // workspace layout
    const size_t MU_ELEMS = (size_t)BN * HID;       // 5,120,000
    float* ws   = (float*)d_ws;
    float* mu0  = ws;
    float* mu1  = mu0 + MU_ELEMS;
    float* deg  = mu1 + MU_ELEMS;                   // BN
    float* spos = deg + BN;
    float* sneg = spos + BN;
    float* uv   = sneg + BN;                        // 6*128: u0,v0,u1,v1,u2,v2
    float* smu  = uv + 6 * HID;                     // B*HID
    float* cb   = smu + BB * HID;                   // B

    // 1) zero the accumulated-scalar region (deg..cb contiguous)
    const int nz = 3 * BN + 6 * HID + BB * HID + BB;
    zero_kernel<<<(nz + 255) / 256, 256, 0, stream>>>(deg, nz);

    // 2) edge pass: deg / s_pos / s_neg
    edge_scal_kernel<<<(BB * EE + 255) / 256, 256, 0, stream>>>(
        edge_ix, edge_w, deg, spos, sneg);

    // 3) per-layer u,v vectors
    uv_kernel<<<1, HID, 0, stream>>>(W3_0, W4_0, uv + 0 * HID, uv + 1 * HID);
    uv_kernel<<<1, HID, 0, stream>>>(W3_r,              W4_r,       uv + 2 * HID, uv + 3 * HID);
    uv_kernel<<<1, HID, 0, stream>>>(W3_r + HID * HID,  W4_r + HID, uv + 4 * HID, uv + 5 * HID);

    // 4) three S2V layers (WMMA f32)
    s2v_layer_kernel<<<NTILES, 256, 0, stream>>>(
        mu_in, IN_DIM, mu0, W2_0, W1_0, uv + 0 * HID, uv + 1 * HID,
        x, deg, spos, sneg);
    s2v_layer_kernel<<<NTILES, 256, 0, stream>>>(
        mu0, HID, mu1, W2_r, W1_r, uv + 2 * HID, uv + 3 * HID,
        x, deg, spos, sneg);
    s2v_layer_kernel<<<NTILES, 256, 0, stream>>>(
        mu1, HID, mu0, W2_r + HID * HID, W1_r + HID, uv + 4 * HID, uv + 5 * HID,
        x, deg, spos, sneg);

    // 5) graph pooling scalar c[b]
    rowsum_kernel<<<dim3(32, BB), 256, 0, stream>>>(mu0, smu);
    cb_kernel<<<BB, HID, 0, stream>>>(smu, W6, W5, cb);

    // 6) readout (WMMA f32 + LDS atomic reduce)
    readout_kernel<<<NTILES, 256, 0, stream>>>(mu0, W7, W5, cb, out);
}